// RPN_3D_loss_smp_78469052498703
// MI455X (gfx1250) — compile-verified
//
#include <hip/hip_runtime.h>
#include <hip/hip_bf16.h>

typedef __attribute__((ext_vector_type(2))) float v2f;
typedef __attribute__((ext_vector_type(8))) float v8f;

#define NACC 7   // fg_cnt, bg_cnt, ce_fg, ce_bg, sl1_2d, sl1_3d, (1-iou)

__device__ __forceinline__ float sl1(float d) {
  float a = fabsf(d);
  return a < 1.f ? 0.5f * a * a : a - 0.5f;
}

// Reduce NACC per-thread accumulators across a 256-thread block.
// Stage 1: wave32 shfl_xor tree -> lane0 of each of 8 waves.
// Stage 2: one V_WMMA_F32_16X16X4_F32 row-sum against an all-ones B matrix
//          (A 16x4: rows j / j+8 hold accumulator j of waves 0-3 / 4-7).
// Returns true on thread 0, with out[0..6] valid there.
__device__ __forceinline__ bool block_reduce7(float acc[NACC], float out[NACC]) {
  __shared__ float smem[64];
  const int lane = threadIdx.x & 31;
  const int wave = threadIdx.x >> 5;
#pragma unroll
  for (int j = 0; j < NACC; ++j) {
#pragma unroll
    for (int off = 16; off; off >>= 1)
      acc[j] += __shfl_xor(acc[j], off, 32);
  }
  if (lane == 0) {
#pragma unroll
    for (int j = 0; j < NACC; ++j) smem[wave * 8 + j] = acc[j];
  }
  __syncthreads();
  if (threadIdx.x < 32) {   // wave 0 only: EXEC all-ones inside (uniform branch)
    const int m     = lane & 15;
    const int kbase = (lane >> 4) ? 2 : 0;  // lanes 16-31 hold K=2,3
    const int row   = m & 7;
    const int wsel  = (m >> 3) ? 4 : 0;     // rows 8-15 cover waves 4-7
    v2f a; a[0] = 0.f; a[1] = 0.f;
    if (row < NACC) {
      a[0] = smem[(kbase + wsel) * 8 + row];
      a[1] = smem[(kbase + 1 + wsel) * 8 + row];
    }
    v2f bones; bones[0] = 1.f; bones[1] = 1.f;  // all-ones B: layout-independent
    v8f c = {};
    v8f d = __builtin_amdgcn_wmma_f32_16x16x4_f32(
        false, a, false, bones, (short)0, c, false, false);
    // D[j][0] on lane0 = rowsum(j); D[j+8][0] on lane16 = rowsum(j+8)
#pragma unroll
    for (int j = 0; j < NACC; ++j) {
      float hi = __shfl(d[j], 16, 32);
      out[j] = d[j] + hi;   // valid on lane 0
    }
    return lane == 0;
  }
  return false;
}

__global__ __launch_bounds__(256) void rpn_partial(
    const float* __restrict__ cls, const float* __restrict__ b2d,
    const float* __restrict__ b3d, const float* __restrict__ b2dt,
    const float* __restrict__ b3dt, const float* __restrict__ rois,
    const float* __restrict__ means, const float* __restrict__ stds,
    const int* __restrict__ labels, const unsigned char* __restrict__ anyv,
    float* __restrict__ parts, int N, int Aanch) {
  const float mn0 = means[0], mn1 = means[1], mn2 = means[2], mn3 = means[3];
  const float sd0 = stds[0],  sd1 = stds[1],  sd2 = stds[2],  sd3 = stds[3];

  float acc[NACC] = {0.f, 0.f, 0.f, 0.f, 0.f, 0.f, 0.f};
  const size_t stride = (size_t)gridDim.x * blockDim.x;

  for (size_t i = (size_t)blockIdx.x * blockDim.x + threadIdx.x; i < (size_t)N;
       i += stride) {
    // prefetch next grid-stride chunk of the hot streams (global_prefetch_b8)
    __builtin_prefetch(cls + (i + stride) * 4, 0, 3);
    __builtin_prefetch(labels + (i + stride), 0, 3);

    const int lab = labels[i];
    const unsigned b = (unsigned)(i / (unsigned)Aanch);
    const bool valid = anyv[b] != 0;
    const bool fg = valid && (lab > 0) && (lab != 3000);
    const bool bg = valid && (lab == 0);

    const float4 cv = ((const float4*)cls)[i];
    if (fg || bg) {
      float mx = fmaxf(fmaxf(cv.x, cv.y), fmaxf(cv.z, cv.w));
      float se = __expf(cv.x - mx) + __expf(cv.y - mx) +
                 __expf(cv.z - mx) + __expf(cv.w - mx);
      int lc = lab < 0 ? 0 : (lab > 3 ? 3 : lab);
      float cl = (lc == 0) ? cv.x : (lc == 1) ? cv.y : (lc == 2) ? cv.z : cv.w;
      float ce = __logf(se) + mx - cl;
      if (fg) { acc[0] += 1.f; acc[2] += ce; }
      else    { acc[1] += 1.f; acc[3] += ce; }
    }

    if (fg) {   // ~2% of lanes: bbox streams only touched here
      const float r0 = rois[i * 5 + 0], r1 = rois[i * 5 + 1];
      const float r2 = rois[i * 5 + 2], r3 = rois[i * 5 + 3];
      const float w = r2 - r0 + 1.f, h = r3 - r1 + 1.f;
      const float cx = r0 + 0.5f * w, cy = r1 + 0.5f * h;

      const float4 d2 = ((const float4*)b2d)[i];
      const float4 t2 = ((const float4*)b2dt)[i];

      // pred box
      float dx = d2.x * sd0 + mn0, dy = d2.y * sd1 + mn1;
      float dw = d2.z * sd2 + mn2, dh = d2.w * sd3 + mn3;
      float pcx = dx * w + cx, pcy = dy * h + cy;
      float pw = __expf(dw) * w, ph = __expf(dh) * h;
      float ax1 = pcx - 0.5f * pw, ay1 = pcy - 0.5f * ph;
      float ax2 = pcx + 0.5f * pw - 1.f, ay2 = pcy + 0.5f * ph - 1.f;
      // target box
      dx = t2.x * sd0 + mn0; dy = t2.y * sd1 + mn1;
      dw = t2.z * sd2 + mn2; dh = t2.w * sd3 + mn3;
      pcx = dx * w + cx; pcy = dy * h + cy;
      pw = __expf(dw) * w; ph = __expf(dh) * h;
      float bx1 = pcx - 0.5f * pw, by1 = pcy - 0.5f * ph;
      float bx2 = pcx + 0.5f * pw - 1.f, by2 = pcy + 0.5f * ph - 1.f;

      float ix1 = fmaxf(ax1, bx1), iy1 = fmaxf(ay1, by1);
      float ix2 = fminf(ax2, bx2), iy2 = fminf(ay2, by2);
      float inter = fmaxf(ix2 - ix1 + 1.f, 0.f) * fmaxf(iy2 - iy1 + 1.f, 0.f);
      float aa = (ax2 - ax1 + 1.f) * (ay2 - ay1 + 1.f);
      float ab = (bx2 - bx1 + 1.f) * (by2 - by1 + 1.f);
      acc[6] += 1.f - inter / (aa + ab - inter);

      acc[4] += sl1(d2.x - t2.x) + sl1(d2.y - t2.y) +
                sl1(d2.z - t2.z) + sl1(d2.w - t2.w);

      const float* p3 = b3d + i * 7;
      const float* q3 = b3dt + i * 7;
      float s3 = 0.f;
#pragma unroll
      for (int k = 0; k < 7; ++k) s3 += sl1(p3[k] - q3[k]);
      acc[5] += s3;
    }
  }

  float tot[NACC];
  bool lead = block_reduce7(acc, tot);
  if (lead) {
#pragma unroll
    for (int j = 0; j < NACC; ++j) parts[(size_t)blockIdx.x * 8 + j] = tot[j];
  }
}

__global__ __launch_bounds__(256) void rpn_final(const float* __restrict__ parts,
                                                 int nparts,
                                                 float* __restrict__ out) {
  float acc[NACC] = {0.f, 0.f, 0.f, 0.f, 0.f, 0.f, 0.f};
  for (int i = threadIdx.x; i < nparts; i += 256) {
#pragma unroll
    for (int j = 0; j < NACC; ++j) acc[j] += parts[(size_t)i * 8 + j];
  }
  float tot[NACC];
  bool lead = block_reduce7(acc, tot);
  if (lead) {
    const float fgc = tot[0], bgc = tot[1];
    // FG_FRACTION/(1-FG_FRACTION) = 0.25
    const float fgw = fgc > 0.f ? 0.25f * bgc / fmaxf(fgc, 1.f) : 0.f;
    const float nact = fmaxf(bgc + (fgw > 0.f ? fgc : 0.f), 1.f);
    const float loss_cls = (tot[2] * fgw + tot[3]) / nact;
    const float den = fmaxf(fgc, 1.f);
    out[0] = loss_cls + (tot[4] + tot[5] + tot[6]) / den;
  }
}

extern "C" void kernel_launch(void* const* d_in, const int* in_sizes, int n_in,
                              void* d_out, int out_size, void* d_ws, size_t ws_size,
                              hipStream_t stream) {
  const float* cls  = (const float*)d_in[0];
  // d_in[1] prob: dead in the loss
  const float* b2d  = (const float*)d_in[2];
  const float* b3d  = (const float*)d_in[3];
  const float* b2dt = (const float*)d_in[4];
  const float* b3dt = (const float*)d_in[5];
  const float* rois = (const float*)d_in[6];
  // d_in[7] anchors: only feeds deleted tensors -> dead
  const float* means = (const float*)d_in[8];
  const float* stds  = (const float*)d_in[9];
  const int* labels  = (const int*)d_in[10];
  // d_in[11..13] labels_fg/bg/ign: derived from labels on device
  const unsigned char* anyv = (const unsigned char*)d_in[14];

  const int N  = in_sizes[10];        // B*A
  const int Bn = in_sizes[14] > 0 ? in_sizes[14] : 1;
  const int Aanch = N / Bn;

  float* parts = (float*)d_ws;
  int grid = (N + 256 * 4 - 1) / (256 * 4);      // ~4 elements/thread
  int maxParts = (int)(ws_size / (8 * sizeof(float)));
  if (grid > maxParts) grid = maxParts;
  if (grid < 1) grid = 1;

  rpn_partial<<<grid, 256, 0, stream>>>(cls, b2d, b3d, b2dt, b3dt, rois,
                                        means, stds, labels, anyv,
                                        parts, N, Aanch);
  rpn_final<<<1, 256, 0, stream>>>(parts, grid, (float*)d_out);
}